// CFConv_2774548873969
// MI455X (gfx1250) — compile-verified
//
#include <hip/hip_runtime.h>
#include <hip/hip_bf16.h>

// CFConv fused for gfx1250 (MI455X): fp32 WMMA (V_WMMA_F32_16X16X4_F32),
// filter network fully fused in LDS so the 128MB W tensor never hits HBM.
// k-outer / 4-accumulator GEMM structure: one B fragment feeds 4 WMMAs.
// Epilogue flags are template params; ssp uses v_exp_f32/v_log_f32.

typedef __attribute__((ext_vector_type(2))) float v2f;
typedef __attribute__((ext_vector_type(8))) float v8f;

#define NB 16      // batch
#define NA 256     // atoms
#define NN 64      // neighbors
#define NG 64      // gaussians
#define NF 128     // filters
#define H_STRIDE 132   // padded LDS row stride for 64x128 intermediate
#define F_STRIDE 68    // padded LDS row stride for 64x64 f_ij tile

__device__ __forceinline__ float ssp_f(float x) {
    // shifted softplus: softplus(x) - ln2, stable, hardware transcendentals.
    // 1 + exp(-|x|) is in (1,2] so v_log_f32 precision is fine.
    return fmaxf(x, 0.0f) + __logf(1.0f + __expf(-fabsf(x))) - 0.69314718055994531f;
}

__device__ __forceinline__ v8f wmma_k4(v2f a, v2f b, v8f c) {
    // D = A(16x4,f32) * B(4x16,f32) + C(16x16,f32)
    return __builtin_amdgcn_wmma_f32_16x16x4_f32(
        /*neg_a=*/false, a, /*neg_b=*/false, b,
        /*c_mod=*/(short)0, c, /*reuse_a=*/false, /*reuse_b=*/false);
}

// M x 128 @ 128 x 128 GEMM, compile-time bias + shifted-softplus options.
// One wave32 computes a 64x16 output strip (4 row-tiles, shared B frags).
template <bool HAS_BIAS, bool DO_SSP>
__global__ __launch_bounds__(256) void gemm128_kernel(
    const float* __restrict__ Am,   // [M,128] row-major
    const float* __restrict__ Bm,   // [128,128] row-major
    const float* __restrict__ bias, // [128] (unused if !HAS_BIAS)
    float* __restrict__ out,        // [M,128]
    int M)
{
    const int wave = threadIdx.x >> 5;
    const int lane = threadIdx.x & 31;
    const int strip = blockIdx.x * 8 + wave;   // 64x16 strips
    const int gid = strip >> 3;                // row-strip index (64 rows)
    const int nt  = strip & 7;
    if (gid * 64 >= M) return;

    const int m     = lane & 15;    // A row within tile / output column
    const int khalf = lane >> 4;    // lane-half: K offset (+2) / M offset (+8)
    const int col   = nt * 16 + m;

    const float* abase = Am + (gid * 64 + m) * 128;
    v8f acc[4] = {{}, {}, {}, {}};
    for (int k0 = 0; k0 < 128; k0 += 4) {
        const int k = k0 + 2 * khalf;
        v2f b; b.x = Bm[k * 128 + col]; b.y = Bm[(k + 1) * 128 + col];
        #pragma unroll
        for (int mt = 0; mt < 4; ++mt) {
            const float* arow = abase + mt * 16 * 128;
            v2f a; a.x = arow[k]; a.y = arow[k + 1];
            acc[mt] = wmma_k4(a, b, acc[mt]);
        }
    }
    const float bv = HAS_BIAS ? bias[col] : 0.0f;
    #pragma unroll
    for (int mt = 0; mt < 4; ++mt) {
        #pragma unroll
        for (int r = 0; r < 8; ++r) {
            const int row = gid * 64 + mt * 16 + r + 8 * khalf;
            float v = acc[mt][r] + bv;
            if (DO_SSP) v = ssp_f(v);
            out[row * 128 + col] = v;
        }
    }
}

// One block per (b,a) pair: filter network on the 64-neighbor tile, cutoff,
// neighbor-feature gather, masked sum over neighbors. 256 threads = 8 wave32.
__global__ __launch_bounds__(256) void cfconv_fused_kernel(
    const float* __restrict__ f_ij,      // [B,A,N,G]
    const float* __restrict__ r_ij,      // [B,A,N]
    const int*   __restrict__ neighbors, // [B,A,N]
    const float* __restrict__ mask,      // [B,A,N]
    const float* __restrict__ fw1,       // [G=64, F=128]
    const float* __restrict__ fb1,       // [128]
    const float* __restrict__ fw2,       // [128,128]
    const float* __restrict__ fb2,       // [128]
    const float* __restrict__ y_in,      // [B*A, 128] = x @ w_in2f
    float* __restrict__ y_agg)           // [B*A, 128]
{
    __shared__ float Flds[NN * F_STRIDE];   // staged f_ij tile (64x64, padded)
    __shared__ float Hlds[NN * H_STRIDE];   // ssp(f_ij@fw1+b1)   (64x128, padded)
    __shared__ float s_lds[NN];             // cutoff * mask per neighbor row
    __shared__ int   nb_lds[NN];            // neighbor indices

    const int bid   = blockIdx.x;           // b*NA + a
    const int bbase = (bid >> 8) << 8;      // b*NA (NA==256)
    const int tid   = threadIdx.x;
    const int wave  = tid >> 5;
    const int lane  = tid & 31;
    const int m     = lane & 15;
    const int khalf = lane >> 4;
    const int col   = wave * 16 + m;        // this wave's output column

    // ---- stage: cutoff*mask, neighbor idx, coalesced f_ij tile -> LDS ----
    if (tid < NN) {
        const float r = r_ij[bid * NN + tid];
        float c = 0.5f * (__cosf(r * 0.62831853071795865f) + 1.0f); // pi/5
        c = (r < 5.0f) ? c : 0.0f;
        s_lds[tid]  = c * mask[bid * NN + tid];
        nb_lds[tid] = neighbors[bid * NN + tid];
    }
    {
        const float4* src = (const float4*)(f_ij + (size_t)bid * NN * NG);
        #pragma unroll
        for (int i = 0; i < 4; ++i) {           // 4096 floats = 1024 float4
            const int v   = i * 256 + tid;
            const float4 d = src[v];
            const int e   = v * 4;
            const int row = e >> 6, c0 = e & 63;
            Flds[row * F_STRIDE + c0 + 0] = d.x;
            Flds[row * F_STRIDE + c0 + 1] = d.y;
            Flds[row * F_STRIDE + c0 + 2] = d.z;
            Flds[row * F_STRIDE + c0 + 3] = d.w;
        }
    }
    __syncthreads();

    // ---- GEMM1: H = ssp(f_tile(64x64) @ fw1(64x128) + fb1) -> LDS ----
    {
        v8f acc[4] = {{}, {}, {}, {}};
        for (int k0 = 0; k0 < NG; k0 += 4) {
            const int k = k0 + 2 * khalf;
            v2f b; b.x = fw1[k * 128 + col]; b.y = fw1[(k + 1) * 128 + col];
            #pragma unroll
            for (int mt = 0; mt < 4; ++mt) {
                const float* arow = Flds + (mt * 16 + m) * F_STRIDE;
                v2f a; a.x = arow[k]; a.y = arow[k + 1];
                acc[mt] = wmma_k4(a, b, acc[mt]);
            }
        }
        const float b1 = fb1[col];
        #pragma unroll
        for (int mt = 0; mt < 4; ++mt) {
            #pragma unroll
            for (int r = 0; r < 8; ++r) {
                const int row = mt * 16 + r + 8 * khalf;
                Hlds[row * H_STRIDE + col] = ssp_f(acc[mt][r] + b1);
            }
        }
    }
    __syncthreads();

    // ---- GEMM2 + cutoff scale + neighbor gather + reduce over N ----
    float partial = 0.0f;
    {
        v8f acc[4] = {{}, {}, {}, {}};
        for (int k0 = 0; k0 < NF; k0 += 4) {
            const int k = k0 + 2 * khalf;
            v2f b; b.x = fw2[k * 128 + col]; b.y = fw2[(k + 1) * 128 + col];
            #pragma unroll
            for (int mt = 0; mt < 4; ++mt) {
                const float* arow = Hlds + (mt * 16 + m) * H_STRIDE;
                v2f a; a.x = arow[k]; a.y = arow[k + 1];
                acc[mt] = wmma_k4(a, b, acc[mt]);
            }
        }
        const float b2 = fb2[col];
        #pragma unroll
        for (int mt = 0; mt < 4; ++mt) {
            #pragma unroll
            for (int r = 0; r < 8; ++r) {
                const int row = mt * 16 + r + 8 * khalf;   // neighbor index n
                const float wv = (acc[mt][r] + b2) * s_lds[row];
                const float yv = y_in[(size_t)(bbase + nb_lds[row]) * 128 + col];
                partial = fmaf(wv, yv, partial);
            }
        }
    }
    // each lane summed 32 of the 64 neighbor rows; other half is in lane^16
    partial += __shfl_xor(partial, 16, 32);
    if (lane < 16) y_agg[(size_t)bid * 128 + col] = partial;
}

extern "C" void kernel_launch(void* const* d_in, const int* in_sizes, int n_in,
                              void* d_out, int out_size, void* d_ws, size_t ws_size,
                              hipStream_t stream) {
    const float* x         = (const float*)d_in[0];
    const float* r_ij      = (const float*)d_in[1];
    const float* f_ij      = (const float*)d_in[2];
    const int*   neighbors = (const int*)  d_in[3];
    const float* pmask     = (const float*)d_in[4];
    const float* fw1       = (const float*)d_in[5];
    const float* fb1       = (const float*)d_in[6];
    const float* fw2       = (const float*)d_in[7];
    const float* fb2       = (const float*)d_in[8];
    const float* w_in2f    = (const float*)d_in[9];
    const float* w_f2out   = (const float*)d_in[10];
    const float* b_f2out   = (const float*)d_in[11];
    float* out = (float*)d_out;

    const int M = NB * NA;               // 4096 rows
    float* y  = (float*)d_ws;            // [4096,128]
    float* y2 = y + (size_t)M * NF;      // [4096,128]

    // 4096/64 = 64 row strips * 8 col tiles = 512 waves = 64 blocks
    const int gemm_blocks = (M / 64) * 8 / 8;

    // y = x @ w_in2f        (no bias, no activation)
    gemm128_kernel<false, false><<<gemm_blocks, 256, 0, stream>>>(x, w_in2f, nullptr, y, M);
    // fused filter network + cutoff + gather + aggregate
    cfconv_fused_kernel<<<NB * NA, 256, 0, stream>>>(
        f_ij, r_ij, neighbors, pmask, fw1, fb1, fw2, fb2, y, y2);
    // out = ssp(y2 @ w_f2out + b_f2out)
    gemm128_kernel<true, true><<<gemm_blocks, 256, 0, stream>>>(y2, w_f2out, b_f2out, out, M);
}